// ReconstructionModule_67508295958904
// MI455X (gfx1250) — compile-verified
//
#include <hip/hip_runtime.h>
#include <hip/hip_bf16.h>

#define B_ 64
#define N_ 576
#define D_ 768

typedef __attribute__((ext_vector_type(2))) float v2f;
typedef __attribute__((ext_vector_type(8))) float v8f;

// ---------------------------------------------------------------------------
// Kernel 1: per-column online softmax-max + argmax over axis=1.
// One thread per (b, j) column. Streams 85 MB once; builds inverse scatter map.
// ---------------------------------------------------------------------------
__global__ __launch_bounds__(256) void posconf_kernel(
    const float* __restrict__ logits, float* __restrict__ conf,
    int* __restrict__ inv)
{
    int tid = blockIdx.x * blockDim.x + threadIdx.x;   // 0 .. B*N-1
    int b = tid / N_;
    int j = tid - b * N_;
    const float* col = logits + (size_t)b * N_ * N_ + j;

    float m = -__builtin_inff();
    float s = 0.0f;
    int arg = 0;
    for (int i = 0; i < N_; ++i) {
        float x = col[(size_t)i * N_];
        if (i + 16 < N_)
            __builtin_prefetch(col + (size_t)(i + 16) * N_, 0, 1);  // global_prefetch_b8
        if (x > m) {                       // strict > keeps first occurrence (JAX argmax)
            s = s * __expf(m - x) + 1.0f;  // rescale running sum to new max
            m = x;
            arg = i;
        } else {
            s += __expf(x - m);
        }
    }
    // max(softmax) = exp(m - m) / sum(exp(x - m)) = 1 / s
    conf[b * N_ + j] = 1.0f / s;
    // inverse scatter map: dest position 'arg' receives source patch 'j'.
    // atomicMax is commutative -> deterministic duplicate resolution.
    atomicMax(&inv[b * N_ + arg], j);
}

// ---------------------------------------------------------------------------
// Branchless 3-tap stencil weight (identity row at endpoints n=0, n=N-1).
// Pure value-selects -> v_cndmask, no EXEC manipulation.
// ---------------------------------------------------------------------------
__device__ __forceinline__ float stencil_w(int n, int kg) {
    const int  dk     = kg - n;
    const bool inband = (unsigned)(dk + 1) <= 2u;          // dk in [-1, +1]
    const bool ep     = (n == 0) | (n == N_ - 1);
    const float w_mid = inband    ? (1.0f / 3.0f) : 0.0f;
    const float w_end = (dk == 0) ? 1.0f          : 0.0f;
    return ep ? w_end : w_mid;
}

// ---------------------------------------------------------------------------
// Branchless gather of rearranged[k][d]: clamp addresses, always load,
// select-zero on invalid k or empty destination slot.
// ---------------------------------------------------------------------------
__device__ __forceinline__ float gather_row(const float* __restrict__ fb,
                                            const int* __restrict__ invb,
                                            int k, int d) {
    int   kc    = min(max(k, 0), N_ - 1);          // safe inv index
    int   s     = invb[kc];
    int   sc    = max(s, 0);                       // safe feature row
    float v     = fb[(size_t)sc * D_ + d];
    bool  valid = ((unsigned)k < (unsigned)N_) & (s >= 0);
    return valid ? v : 0.0f;
}

// ---------------------------------------------------------------------------
// Kernel 2: fused gather (via inv) + 3-tap smooth (banded matmul on the WMMA
// pipe) + [B,N,D] -> [B,D,N] transpose. One wave per 16x16 (n,d) output tile.
//
// D = S(16xK) x R(Kx16) accumulated over 5 K=4 chunks covering n0-2..n0+17.
// A-matrix (f32 16x4): lanes 0-15 hold rows M, VGPR0=K0,VGPR1=K1;
//                      lanes 16-31 VGPR0=K2,VGPR1=K3.
// D-matrix: lane = d column, VGPR r = consecutive n -> transposed store is
// two contiguous float4 stores per lane.
// ---------------------------------------------------------------------------
__global__ __launch_bounds__(256) void smooth_wmma_kernel(
    const float* __restrict__ features, const int* __restrict__ inv,
    float* __restrict__ out)
{
    const int lane = threadIdx.x & 31;
    const int wave = threadIdx.x >> 5;
    const int tilesD = D_ / 16;                     // 48
    const int tilesPerB = (N_ / 16) * tilesD;       // 36*48 = 1728
    int tile = blockIdx.x * (blockDim.x >> 5) + wave;
    int b  = tile / tilesPerB;
    int t  = tile - b * tilesPerB;
    int tn = t / tilesD;
    int n0 = tn * 16;
    int d0 = (t - tn * tilesD) * 16;

    const int half = lane >> 4;       // 0: K0/K1 via lanes 0-15; 1: K2/K3
    const int l15  = lane & 15;
    const int n = n0 + l15;           // A-matrix row (output patch index)
    const int d = d0 + l15;           // B-matrix column (feature index)
    const int*   invb = inv + b * N_;
    const float* fb   = features + (size_t)b * N_ * D_;

    v8f c = {};
#pragma unroll
    for (int ch = 0; ch < 5; ++ch) {
        int k0 = n0 - 2 + 4 * ch + half * 2;     // this lane's two K values
        v2f a, bm;
        a.x  = stencil_w(n, k0);
        a.y  = stencil_w(n, k0 + 1);
        bm.x = gather_row(fb, invb, k0,     d);
        bm.y = gather_row(fb, invb, k0 + 1, d);
        // (neg_a, A, neg_b, B, c_mod, C, reuse_a, reuse_b)
        c = __builtin_amdgcn_wmma_f32_16x16x4_f32(
                false, a, false, bm, (short)0, c, false, false);
    }

    // Transposed store: out[b][d][n0 + half*8 + r] = c[r]
    float* op = out + ((size_t)b * D_ + d) * N_ + n0 + half * 8;
    *(float4*)(op)     = make_float4(c[0], c[1], c[2], c[3]);
    *(float4*)(op + 4) = make_float4(c[4], c[5], c[6], c[7]);
}

// ---------------------------------------------------------------------------
extern "C" void kernel_launch(void* const* d_in, const int* in_sizes, int n_in,
                              void* d_out, int out_size, void* d_ws, size_t ws_size,
                              hipStream_t stream) {
    const float* features = (const float*)d_in[0];   // [B, N, D]
    const float* logits   = (const float*)d_in[1];   // [B, N, N]
    float* out  = (float*)d_out;                     // [B, D, N] then [B, N] conf
    float* conf = out + (size_t)B_ * D_ * N_;
    int*   inv  = (int*)d_ws;                        // [B, N] inverse scatter map

    // init inverse map to -1 (0xFFFFFFFF); memset is graph-capturable
    hipMemsetAsync(inv, 0xFF, (size_t)B_ * N_ * sizeof(int), stream);

    int cols = B_ * N_;                              // 36864 columns
    posconf_kernel<<<cols / 256, 256, 0, stream>>>(logits, conf, inv);

    int tiles = B_ * (N_ / 16) * (D_ / 16);          // 110592 tiles, 8 waves/block
    smooth_wmma_kernel<<<tiles / 8, 256, 0, stream>>>(features, inv, out);
}